// GELU129_39857296507315
// MI455X (gfx1250) — compile-verified
//
#include <hip/hip_runtime.h>
#include <math.h>

// ---------------------------------------------------------------------------
// Problem constants (from reference): x is (B,T,D) fp32, scan is along T.
// ---------------------------------------------------------------------------
constexpr int Bdim = 4;
constexpr int Tdim = 4096;
constexpr int Ddim = 2048;
constexpr int NC   = 64;          // chunks along T (parallel scan)
constexpr int LCH  = Tdim / NC;   // 64 steps per chunk

constexpr float EPSF    = 1e-5f;
constexpr float EPSVARF = 1e-4f;
constexpr float GELU_CF = 0.7978845608028654f;   // sqrt(2/pi)

// Workspace layout (float offsets). Total ~4.2 MB.
constexpr size_t WS_PAR   = 0;                     // 16 floats of derived scalars
constexpr size_t WS_CMEAN = 16;                    // D floats
constexpr size_t WS_CINV  = 16 + (size_t)Ddim;     // D floats
constexpr size_t WS_RESM  = 16 + 2 * (size_t)Ddim; // B*NC*D floats
constexpr size_t WS_RESS  = WS_RESM + (size_t)Bdim * NC * Ddim;

#define USE_ASYNC_LDS 1

// ---------------------------------------------------------------------------
// gfx1250 helpers
// ---------------------------------------------------------------------------
__device__ __forceinline__ float ftanh(float v) {
#if __has_builtin(__builtin_amdgcn_tanhf)
    return __builtin_amdgcn_tanhf(v);          // v_tanh_f32 (CDNA5 trans op)
#else
    float r;
    asm volatile("v_tanh_f32 %0, %1\n\tv_nop" : "=v"(r) : "v"(v));
    return r;
#endif
}

__device__ __forceinline__ void wait_async0() {
#if __has_builtin(__builtin_amdgcn_s_wait_asynccnt)
    __builtin_amdgcn_s_wait_asynccnt(0);
#else
    asm volatile("s_wait_asynccnt 0x0" ::: "memory");
#endif
}

// GLOBAL_LOAD_ASYNC_TO_LDS_B128: VDST = VGPR holding LDS byte address,
// VADDR = 64-bit global address. Deriving the LDS offset from the real
// element pointer (generic LDS pointers keep the LDS offset in addr[31:0])
// both gives the exact offset and forces the LDS array to escape into the
// asm, so LLVM cannot fold the later tile[] loads to poison.
__device__ __forceinline__ void async_b128_to_lds(const float* gptr, float* lptr) {
    const unsigned lds_off = (unsigned)(uintptr_t)lptr;
    asm volatile("global_load_async_to_lds_b128 %0, %1, off"
                 :: "v"(lds_off),
                    "v"((unsigned long long)(uintptr_t)gptr)
                 : "memory");
}

// ---------------------------------------------------------------------------
// Pass 3 (defined first so the disasm snippet shows the CDNA5 paths):
// stage the 64x256 fp32 tile through LDS with async b128 copies, then run the
// exact scan from the chunk carry and emit gelu(x * gate) with NT stores.
// 64 KB LDS/block -> 5 blocks (40 wave32s) per 320 KB WGP.
// ---------------------------------------------------------------------------
__global__ __launch_bounds__(256) void gate_main_kernel(const float* __restrict__ x,
                                                        const float* __restrict__ ws,
                                                        float* __restrict__ out) {
    __shared__ float tile[LCH * 256];   // 64 KB

    const int tid    = threadIdx.x;
    const int chBase = blockIdx.x * 256;
    const int c      = blockIdx.y;
    const int b      = blockIdx.z;
    const int d      = chBase + tid;
    const size_t rowBase = ((size_t)b * Tdim + (size_t)c * LCH) * Ddim + chBase;

    // ---- stage tile: 16 async b128 per thread (ASYNCcnt += 16 per wave) ----
    {
        const int sub  = tid & 63;    // 4-channel group within the 256-ch tile
        const int row4 = tid >> 6;    // 0..3
#if USE_ASYNC_LDS
#pragma unroll
        for (int k = 0; k < LCH / 4; ++k) {
            const int r = k * 4 + row4;
            async_b128_to_lds(x + rowBase + (size_t)r * Ddim + sub * 4,
                              &tile[r * 256 + sub * 4]);
        }
        wait_async0();
#else
#pragma unroll
        for (int k = 0; k < LCH / 4; ++k) {
            const int r = k * 4 + row4;
            const float4 v = *(const float4*)(x + rowBase + (size_t)r * Ddim + sub * 4);
            *(float4*)&tile[r * 256 + sub * 4] = v;
        }
#endif
    }
    __syncthreads();

    // ---- uniform scalars + per-channel constants ----
    const float sig_g = ws[WS_PAR + 0];
    const float al_g  = ws[WS_PAR + 1];
    const float dl    = ws[WS_PAR + 2];
    const float sig_l = ws[WS_PAR + 3];
    const float al_l  = ws[WS_PAR + 4];
    const float om    = 1.0f - dl;
    const float cm    = ws[WS_CMEAN + d];
    const float ci    = ws[WS_CINV  + d];

    float m = 0.0f, s = 0.0f;
    if (c > 0) {
        const size_t idx = ((size_t)b * NC + c) * Ddim + d;
        m = ws[WS_RESM + idx];      // exact carry-in from pass 2
        s = ws[WS_RESS + idx];
    }

    float* op = out + rowBase + tid;

#pragma unroll 4
    for (int j = 0; j < LCH; ++j) {
        const float v = tile[j * 256 + tid];   // conflict-free: 32 lanes -> 32 banks

        float surp_l;
        if (c == 0 && j == 0) {
            m = v; s = v * v;                  // t==0: init state, surprise = 0
            surp_l = 0.0f;
        } else {
            const float var = fmaxf(s - m * m, EPSVARF);
            const float z   = (v - m) / (sqrtf(var) + EPSF);
            surp_l = ftanh(sig_l * fabsf(z));
            m = dl * m + om * v;
            s = dl * s + om * (v * v);
        }

        const float zg     = (v - cm) * ci;
        const float surp_g = ftanh(sig_g * fabsf(zg));

        const float gate = 1.0f + al_g * surp_g + al_l * surp_l;
        const float y    = v * gate;
        const float t3   = GELU_CF * (y + 0.044715f * y * y * y);
        const float r    = 0.5f * y * (1.0f + ftanh(t3));
        __builtin_nontemporal_store(r, op + (size_t)j * Ddim);  // out is write-once
    }
}

// ---------------------------------------------------------------------------
// Kernel 0: derive scalar parameters once on device (graph-capture safe).
// Fast HW intrinsics (v_exp_f32 / v_log_f32) instead of scalarized libm.
// ---------------------------------------------------------------------------
__global__ void gate_params_kernel(const float* __restrict__ log_sigma_g,
                                   const float* __restrict__ log_alpha_g,
                                   const float* __restrict__ logit_decay_l,
                                   const float* __restrict__ log_sigma_l,
                                   const float* __restrict__ log_alpha_l,
                                   float* __restrict__ ws) {
    if (threadIdx.x == 0) {
        const float sg = __logf(1.0f + __expf(log_sigma_g[0]));     // softplus
        const float ag = __logf(1.0f + __expf(log_alpha_g[0]));
        const float dl = 1.0f / (1.0f + __expf(-logit_decay_l[0])); // sigmoid
        const float sl = __logf(1.0f + __expf(log_sigma_l[0]));
        const float al = __logf(1.0f + __expf(log_alpha_l[0]));
        ws[WS_PAR + 0] = sg;
        ws[WS_PAR + 1] = ag;
        ws[WS_PAR + 2] = dl;
        ws[WS_PAR + 3] = sl;
        ws[WS_PAR + 4] = al;
        // A = d^LCH = exp2(LCH * log2(d)); v_exp_f32 is natively base-2.
        ws[WS_PAR + 5] = __builtin_amdgcn_exp2f((float)LCH * __log2f(dl));
    }
}

// ---------------------------------------------------------------------------
// Kernel 1: per-channel global z-score constants: mean and 1/(sqrt(var)+eps).
// ---------------------------------------------------------------------------
__global__ void gate_chan_kernel(const float* __restrict__ mean_g,
                                 const float* __restrict__ sq_g,
                                 float* __restrict__ ws) {
    int d = blockIdx.x * blockDim.x + threadIdx.x;
    if (d < Ddim) {
        float m   = mean_g[d];
        float var = fmaxf(sq_g[d] - m * m, EPSVARF);
        ws[WS_CMEAN + d] = m;
        ws[WS_CINV  + d] = 1.0f / (sqrtf(var) + EPSF);
    }
}

// ---------------------------------------------------------------------------
// Pass 1: per-chunk scan summaries.
// EMA is affine: state' = d*state + (1-d)*u, so over a chunk:
//   state_end = d^L * state_start + R,  R = (1-d) * sum_j d^(L-1-j) * u_j.
// Chunk 0 stores the *absolute* end state (init m=x0, s=x0^2).
// 8-deep register double buffer keeps 8 coalesced loads in flight per thread.
// ---------------------------------------------------------------------------
__global__ __launch_bounds__(256) void gate_scan_part1(const float* __restrict__ x,
                                                       float* __restrict__ ws) {
    const int tid = threadIdx.x;
    const int d   = blockIdx.x * 256 + tid;
    const int c   = blockIdx.y;
    const int b   = blockIdx.z;

    const float dl = ws[WS_PAR + 2];
    const float om = 1.0f - dl;
    const float* xp = x + ((size_t)b * Tdim + (size_t)c * LCH) * Ddim + d;

    constexpr int PF = 8;
    float cur[PF], nxt[PF];
#pragma unroll
    for (int u = 0; u < PF; ++u) cur[u] = xp[(size_t)u * Ddim];

    float m = 0.0f, s = 0.0f;
    const bool first_chunk = (c == 0);

    for (int jj = 0; jj < LCH; jj += PF) {
        if (jj + PF < LCH) {
#pragma unroll
            for (int u = 0; u < PF; ++u) nxt[u] = xp[(size_t)(jj + PF + u) * Ddim];
        }
#pragma unroll
        for (int u = 0; u < PF; ++u) {
            float v = cur[u];
            if (first_chunk && jj == 0 && u == 0) {
                m = v; s = v * v;               // scan init at t==0
            } else {
                m = dl * m + om * v;
                s = dl * s + om * v * v;
            }
        }
#pragma unroll
        for (int u = 0; u < PF; ++u) cur[u] = nxt[u];
    }

    const size_t idx = ((size_t)b * NC + c) * Ddim + d;
    ws[WS_RESM + idx] = m;
    ws[WS_RESS + idx] = s;
}

// ---------------------------------------------------------------------------
// Pass 2: serial prefix over the NC=64 chunk summaries, one thread per (b,d).
// In-place: slot c is rewritten with the carry-in state for chunk c.
// ---------------------------------------------------------------------------
__global__ __launch_bounds__(256) void gate_scan_part2(float* __restrict__ ws) {
    const int idx = blockIdx.x * 256 + threadIdx.x;   // b*Ddim + d
    const int b = idx / Ddim;
    const int d = idx - b * Ddim;
    const float A = ws[WS_PAR + 5];

    float* rm = &ws[WS_RESM + (size_t)b * NC * Ddim + d];
    float* rs = &ws[WS_RESS + (size_t)b * NC * Ddim + d];

    float pm = rm[0];   // absolute end state of chunk 0
    float ps = rs[0];
    for (int c = 1; c < NC; ++c) {
        const size_t o = (size_t)c * Ddim;
        const float cmv = rm[o];
        const float csv = rs[o];
        rm[o] = pm;                 // carry-in for chunk c
        rs[o] = ps;
        pm = A * pm + cmv;
        ps = A * ps + csv;
    }
}

// ---------------------------------------------------------------------------
// Host launcher (graph-capture safe: kernels only, all on `stream`).
// ---------------------------------------------------------------------------
extern "C" void kernel_launch(void* const* d_in, const int* in_sizes, int n_in,
                              void* d_out, int out_size, void* d_ws, size_t ws_size,
                              hipStream_t stream) {
    const float* x   = (const float*)d_in[0];   // (B,T,D)
    const float* emg = (const float*)d_in[1];   // ema_mean_g (D)
    const float* esg = (const float*)d_in[2];   // ema_sq_g   (D)
    // d_in[3] = logit_decay_g: only updates EMA buffers in torch, not output.
    const float* lsg = (const float*)d_in[4];
    const float* lag = (const float*)d_in[5];
    const float* ldl = (const float*)d_in[6];
    const float* lsl = (const float*)d_in[7];
    const float* lal = (const float*)d_in[8];

    float* out = (float*)d_out;
    float* ws  = (float*)d_ws;

    gate_params_kernel<<<1, 32, 0, stream>>>(lsg, lag, ldl, lsl, lal, ws);
    gate_chan_kernel<<<Ddim / 256, 256, 0, stream>>>(emg, esg, ws);

    dim3 grid(Ddim / 256, NC, Bdim);            // (8, 64, 4) = 2048 blocks
    gate_scan_part1<<<grid, 256, 0, stream>>>(x, ws);
    gate_scan_part2<<<(Bdim * Ddim) / 256, 256, 0, stream>>>(ws);
    gate_main_kernel<<<grid, 256, 0, stream>>>(x, ws, out);
}